// GCNArchEmbedder_32091995636315
// MI455X (gfx1250) — compile-verified
//
#include <hip/hip_runtime.h>

typedef __attribute__((ext_vector_type(2))) float v2f;
typedef __attribute__((ext_vector_type(8))) float v8f;

#define NUM_STEPS 4
#define OP_DIM    48
#define OP_HID    48
#define GCN0      128
#define GCN1      128
#define NROWS     66          // 2 init rows + 64 op-pair combos
#define BATCH     32768
#define NCG       2
#define NSAMPLES  (BATCH * NCG)
#define SPB       16          // samples (output rows) per block
#define VPAD      132         // padded V row length (stride 4 banks -> conflict free col reads)

// ---------------------------------------------------------------------------
// Kernel 1: build the 66x128 table  Z[r] = (x_row[r] @ x_hidden_w + b) @ gcn_w1
//   r = 0,1          : x_row = init_node_emb[r]            (96-dim)
//   r = 2 + i*8 + j  : x_row = concat(op_emb[i], op_emb[j])
// Tiny (1.4 MFLOP): one block.
// ---------------------------------------------------------------------------
__global__ __launch_bounds__(256) void build_ztab(
    const float* __restrict__ initEmb,   // 2 x 96
    const float* __restrict__ opEmb,     // 8 x 48
    const float* __restrict__ xW,        // 96 x 48
    const float* __restrict__ xB,        // 48
    const float* __restrict__ gW1,       // 48 x 128
    float* __restrict__ ztab)            // 66 x 128 (workspace)
{
    __shared__ float H[NROWS][OP_HID];
    const int tid = threadIdx.x;

    for (int idx = tid; idx < NROWS * OP_HID; idx += 256) {
        const int r = idx / OP_HID;
        const int k = idx - r * OP_HID;
        float acc = xB[k];
        if (r < 2) {
            const float* xr = initEmb + r * 96;
            for (int d = 0; d < 96; ++d) acc += xr[d] * xW[d * OP_HID + k];
        } else {
            const int c = r - 2;
            const float* e0 = opEmb + (c >> 3) * OP_DIM;
            const float* e1 = opEmb + (c & 7) * OP_DIM;
            for (int d = 0; d < 48; ++d) acc += e0[d] * xW[d * OP_HID + k];
            for (int d = 0; d < 48; ++d) acc += e1[d] * xW[(48 + d) * OP_HID + k];
        }
        H[r][k] = acc;
    }
    __syncthreads();
    for (int idx = tid; idx < NROWS * GCN0; idx += 256) {
        const int r = idx >> 7;
        const int f = idx & 127;
        float acc = 0.f;
        for (int k = 0; k < OP_HID; ++k) acc += H[r][k] * gW1[k * GCN0 + f];
        ztab[idx] = acc;
    }
}

// ---------------------------------------------------------------------------
// Kernel 2: per sample s build V[s] = 0.25 * sum_t cnt[t]*relu(Z[p0]+Z[p1]),
// then GEMM  out(16x128 tile) = V(16x128) @ gcn_w2(128x128)
// via V_WMMA_F32_16X16X4_F32 (exact f32 path).
// ---------------------------------------------------------------------------
__global__ __launch_bounds__(256) void gcn_embed_main(
    const int*   __restrict__ archs,   // (BATCH, 2, 2, 8) int32; [.,.,0,:]=prev, [.,.,1,:]=ops
    const float* __restrict__ ztab,    // 66 x 128
    const float* __restrict__ gW2,     // 128 x 128
    float*       __restrict__ out)     // NSAMPLES x 128  (== (BATCH, 256))
{
    __shared__ float Z[NROWS][GCN0];     // 33.8 KB, gathered per f (bank friendly)
    __shared__ float V[SPB][VPAD];       //  8.4 KB
    __shared__ int   eRow[SPB][8];       // Z-row index per edge
    __shared__ float cnt[SPB][4];        // in-degree weight per step node

    const int tid   = threadIdx.x;
    const int sbase = blockIdx.x * SPB;

    // Stage Z table into LDS (L2-hot broadcast)
    for (int idx = tid; idx < NROWS * GCN0; idx += 256)
        (&Z[0][0])[idx] = ztab[idx];

    // Per-sample scalar prep: 16 threads, one sample each
    if (tid < SPB) {
        const int* a = archs + (long)(sbase + tid) * 16;
        int prev[8], ops[8];
        #pragma unroll
        for (int e = 0; e < 8; ++e) { prev[e] = a[e] & 7; ops[e] = a[8 + e] & 7; }
        int zrow[6];
        zrow[0] = 0; zrow[1] = 1;
        #pragma unroll
        for (int t = 0; t < NUM_STEPS; ++t)
            zrow[2 + t] = 2 + ops[2 * t] * 8 + ops[2 * t + 1];
        #pragma unroll
        for (int e = 0; e < 8; ++e) {
            int p = prev[e]; if (p > 5) p = 5;
            eRow[tid][e] = zrow[p];
        }
        #pragma unroll
        for (int t = 0; t < NUM_STEPS; ++t) {
            int c = 0;
            #pragma unroll
            for (int e = 0; e < 8; ++e) c += (prev[e] == t + 2) ? 1 : 0;
            cnt[tid][t] = (float)c;
        }
    }
    __syncthreads();

    // Build V tile: V[s][f] = 0.25 * sum_t cnt[t] * relu(Z[eRow[2t]][f] + Z[eRow[2t+1]][f])
    for (int idx = tid; idx < SPB * GCN0; idx += 256) {
        const int s = idx >> 7;
        const int f = idx & 127;
        float v = 0.f;
        #pragma unroll
        for (int t = 0; t < NUM_STEPS; ++t) {
            float y = Z[eRow[s][2 * t]][f] + Z[eRow[s][2 * t + 1]][f];
            y = y > 0.f ? y : 0.f;
            v += cnt[s][t] * y;
        }
        V[s][f] = 0.25f * v;
    }
    __syncthreads();

    // WMMA GEMM: 8 waves, wave w owns output columns [16w, 16w+16)
    // f32 16x16x4 fragment layouts (ISA 7.12.2):
    //   A: lane(0..15)=row M, v0 holds K=k0+2*hi, v1 holds K=k0+2*hi+1
    //   B: lane(0..15)=col N, v0 holds K=k0+2*hi, v1 holds K=k0+2*hi+1
    //   C/D: vgpr r = row r (lanes 0-15) / row r+8 (lanes 16-31), lane = col
    const int wave    = tid >> 5;
    const int lane    = tid & 31;
    const int hi      = lane >> 4;
    const int lo      = lane & 15;
    const int colbase = wave * 16;

    v8f acc = {};
    #pragma unroll
    for (int k0 = 0; k0 < GCN0; k0 += 4) {
        const int ka = k0 + 2 * hi;
        v2f a; a.x = V[lo][ka];
               a.y = V[lo][ka + 1];
        v2f b; b.x = gW2[ka * GCN1 + colbase + lo];
               b.y = gW2[(ka + 1) * GCN1 + colbase + lo];
        acc = __builtin_amdgcn_wmma_f32_16x16x4_f32(
            /*neg_a=*/false, a, /*neg_b=*/false, b,
            /*c_mod=*/(short)0, acc, /*reuse_a=*/false, /*reuse_b=*/false);
    }

    // Store D tile: row = r + 8*hi, col = colbase + lo
    #pragma unroll
    for (int r = 0; r < 8; ++r) {
        const int row = r + 8 * hi;
        out[(long)(sbase + row) * GCN1 + colbase + lo] = acc[r];
    }
}

// ---------------------------------------------------------------------------
extern "C" void kernel_launch(void* const* d_in, const int* in_sizes, int n_in,
                              void* d_out, int out_size, void* d_ws, size_t ws_size,
                              hipStream_t stream) {
    const int*   archs = (const int*)  d_in[0];
    const float* initE = (const float*)d_in[1];
    const float* opE   = (const float*)d_in[2];
    const float* xW    = (const float*)d_in[3];
    const float* xB    = (const float*)d_in[4];
    const float* gW1   = (const float*)d_in[5];
    const float* gW2   = (const float*)d_in[6];
    float* outp = (float*)d_out;
    float* ztab = (float*)d_ws;   // 66*128*4 = 33792 bytes of scratch

    build_ztab<<<1, 256, 0, stream>>>(initE, opE, xW, xB, gW1, ztab);
    gcn_embed_main<<<NSAMPLES / SPB, 256, 0, stream>>>(archs, ztab, gW2, outp);
}